// TriPlaneEncoder_36910948941984
// MI455X (gfx1250) — compile-verified
//
#include <hip/hip_runtime.h>

typedef float v4f __attribute__((ext_vector_type(4)));

#define RES 1024

// One thread = one (point, level). Block of 256 threads = 64 points x 4 levels.
__global__ __launch_bounds__(256) void triplane_enc_kernel(
    const float* __restrict__ positions,   // [B,3]
    const float* __restrict__ table,       // [3, 1024*1024, 2]
    float* __restrict__ out,               // [B, 32]
    int B)
{
    __shared__ float sp[192];              // 64 points * 3 coords

    const int tid = threadIdx.x;
    const long long blockPt0 = (long long)blockIdx.x * 64;

    // ---- Stage this block's positions into LDS (CDNA5 async global->LDS) ----
#if __has_builtin(__builtin_amdgcn_global_load_async_to_lds_b32)
    if (tid < 192) {
        long long gofs = blockPt0 * 3 + tid;
        if (gofs < (long long)B * 3) {
            __builtin_amdgcn_global_load_async_to_lds_b32(
                (__attribute__((address_space(1))) int*)(positions + gofs),
                (__attribute__((address_space(3))) int*)(&sp[tid]),
                /*offset=*/0, /*cpol=*/0);
        }
    }
  #if __has_builtin(__builtin_amdgcn_s_wait_asynccnt)
    __builtin_amdgcn_s_wait_asynccnt(0);
  #else
    asm volatile("s_wait_asynccnt 0" ::: "memory");
  #endif
#else
    if (tid < 192) {
        long long gofs = blockPt0 * 3 + tid;
        sp[tid] = (gofs < (long long)B * 3) ? positions[gofs] : 0.0f;
    }
#endif
    __syncthreads();

    const int localPt = tid >> 2;          // 0..63
    const int level   = tid & 3;           // 0..3
    const long long pt = blockPt0 + localPt;
    if (pt >= B) return;

    const float p3[3] = { sp[localPt * 3 + 0],
                          sp[localPt * 3 + 1],
                          sp[localPt * 3 + 2] };

    const float s   = (float)((128 << level) - 1);   // scale - 1
    const int   fac = 8 >> level;                    // factor

    float fr[3];
    int ia[3], ib[3];
#pragma unroll
    for (int d = 0; d < 3; ++d) {
        float v  = fmaf(p3[d], s, 0.5f);
        float fl = floorf(v);
        int   g  = (int)fl;
        fr[d]    = v - fl;
        int lo = g * fac;
        int hi = lo + fac;
        ia[d] = lo < (RES - 1) ? lo : (RES - 1);     // min(g*fac, 1023)
        ib[d] = hi < (RES - 1) ? hi : (RES - 1);     // min((g+1)*fac, 1023)
    }

    // Planes use dim pairs (0,1), (1,2), (2,0); idx = i_a + i_b * RES.
    float accx[3], accy[3];
#pragma unroll
    for (int p = 0; p < 3; ++p) {
        const int a = p;
        const int b = (p == 2) ? 0 : p + 1;
        const float2* tp = (const float2*)table + (size_t)p * (RES * RES);

        const int r0 = ia[b] * RES;
        const int r1 = ib[b] * RES;
        // Issue all 4 corner gathers (global_load_b64 each) before combining.
        float2 c00 = tp[ia[a] + r0];
        float2 c10 = tp[ib[a] + r0];
        float2 c01 = tp[ia[a] + r1];
        float2 c11 = tp[ib[a] + r1];

        const float wa1 = fr[a], wa0 = 1.0f - wa1;
        const float wb1 = fr[b], wb0 = 1.0f - wb1;
        const float w00 = wa0 * wb0, w10 = wa1 * wb0;
        const float w01 = wa0 * wb1, w11 = wa1 * wb1;

        accx[p] = w00 * c00.x + w10 * c10.x + w01 * c01.x + w11 * c11.x;
        accy[p] = w00 * c00.y + w10 * c10.y + w01 * c01.y + w11 * c11.y;
    }

    // Output layout per (point, level): [a0x a0y a1x a1y a2x a2y px py]
    const float prx = accx[0] * accx[1] * accx[2];
    const float pry = accy[0] * accy[1] * accy[2];

    v4f o0 = { accx[0], accy[0], accx[1], accy[1] };
    v4f o1 = { accx[2], accy[2], prx,     pry     };

    const long long tgl = (long long)blockIdx.x * 256 + tid;  // == pt*4 + level
    v4f* dst = (v4f*)(out + tgl * 8);
    // Streaming 256MB output: non-temporal so it doesn't evict the hot 24MB
    // table from L2.
    __builtin_nontemporal_store(o0, dst + 0);
    __builtin_nontemporal_store(o1, dst + 1);
}

extern "C" void kernel_launch(void* const* d_in, const int* in_sizes, int n_in,
                              void* d_out, int out_size, void* d_ws, size_t ws_size,
                              hipStream_t stream) {
    const float* positions = (const float*)d_in[0];   // B*3 floats
    const float* table     = (const float*)d_in[1];   // 3*1024*1024*2 floats
    float*       out       = (float*)d_out;           // B*32 floats

    const int B      = in_sizes[0] / 3;
    const int blocks = (B + 63) / 64;                 // 64 points per block

    triplane_enc_kernel<<<blocks, 256, 0, stream>>>(positions, table, out, B);
}